// NGPPlSampling_47141561041220
// MI455X (gfx1250) — compile-verified
//
#include <hip/hip_runtime.h>
#include <cstdint>
#include <cstddef>

// ---------------------------------------------------------------------------
// NGP-style volumetric compositing, MI455X (gfx1250, wave32).
// One wave per ray. Async global->LDS staging of rgbs (CDNA5 async path),
// wave32 shuffle scan for the transmittance cumsum.
// ---------------------------------------------------------------------------

#define NSAMP 256           // samples per ray (reference-fixed)
#define SPL   8             // samples per lane (256 / 32)
#define WPB   8             // waves per 256-thread block
#define LANE_STRIDE_B 112   // 96B of rgb data per lane + 16B pad (bank-conflict relief)
#define WAVE_STRIDE_B (32 * LANE_STRIDE_B)   // 3584 B per wave

__global__ __launch_bounds__(256)
void ngp_render_kernel(const float* __restrict__ rays,
                       const float* __restrict__ sigmas,
                       const float* __restrict__ rgbs,
                       const float* __restrict__ center,
                       const float* __restrict__ half_size,
                       float* __restrict__ out,
                       int n_rays)
{
    __shared__ __align__(16) unsigned char lds_raw[WPB * WAVE_STRIDE_B];

    const int lane = threadIdx.x & 31;
    const int wv   = threadIdx.x >> 5;
    const int ray  = blockIdx.x * WPB + wv;
    if (ray >= n_rays) return;          // wave-uniform guard

    // ---------------- async-stage this ray's rgbs (3072 B) into LDS ----------
    // Instruction i moves a fully-coalesced contiguous 512B slab
    // (lane*16 within the slab). LDS destination is remapped so each lane's
    // 96B of samples lands in a 112B padded region (q = g/96 selects region).
    const uint32_t lds_wave = (uint32_t)(uintptr_t)&lds_raw[0]
                            + (uint32_t)(wv * WAVE_STRIDE_B);
    const uint64_t rgb_base = (uint64_t)(uintptr_t)rgbs;
    const uint32_t ray_off  = (uint32_t)ray * (uint32_t)(NSAMP * 3 * 4); // 3072B
#pragma unroll
    for (int i = 0; i < 6; ++i) {
        const uint32_t g    = (uint32_t)(i * 512) + (uint32_t)(lane * 16);
        const uint32_t q    = g / 96u;            // lane-region index (8 samples)
        const uint32_t rem  = g - q * 96u;        // multiple of 16 (96 = 6*16)
        const uint32_t ldsa = lds_wave + q * (uint32_t)LANE_STRIDE_B + rem;
        const uint32_t voff = ray_off + g;
        asm volatile("global_load_async_to_lds_b128 %0, %1, %2"
                     :: "v"(ldsa), "v"(voff), "s"(rgb_base)
                     : "memory");
    }

    // ---------------- ray / AABB setup (wave-uniform, done in all lanes) -----
    const float* rp = rays + (size_t)ray * 6;
    float ox = rp[0], oy = rp[1], oz = rp[2];
    float dx = rp[3], dy = rp[4], dz = rp[5];
    const float inv_norm = 1.0f / (sqrtf(dx*dx + dy*dy + dz*dz) + 1e-8f);
    dx *= inv_norm; dy *= inv_norm; dz *= inv_norm;

    const float cx = center[0],    cy = center[1],    cz = center[2];
    const float hx = half_size[0], hy = half_size[1], hz = half_size[2];

    const float sdx = (fabsf(dx) > 1e-8f) ? dx : 1e-8f;
    const float sdy = (fabsf(dy) > 1e-8f) ? dy : 1e-8f;
    const float sdz = (fabsf(dz) > 1e-8f) ? dz : 1e-8f;
    const float ix = 1.0f / sdx, iy = 1.0f / sdy, iz = 1.0f / sdz;

    const float t1x = (cx - hx - ox) * ix, t2x = (cx + hx - ox) * ix;
    const float t1y = (cy - hy - oy) * iy, t2y = (cy + hy - oy) * iy;
    const float t1z = (cz - hz - oz) * iz, t2z = (cz + hz - oz) * iz;

    const float tmin = fmaxf(fmaxf(fminf(t1x,t2x), fminf(t1y,t2y)), fminf(t1z,t2z));
    const float tmax = fminf(fminf(fmaxf(t1x,t2x), fmaxf(t1y,t2y)), fmaxf(t1z,t2z));
    const bool  hit  = tmax > fmaxf(tmin, 0.0f);

    float nearr = fmaxf(tmin, 0.0f);
    if (hit && (nearr < 0.01f)) nearr = 0.01f;
    const float farr  = hit ? tmax : nearr;
    const float delta = (farr - nearr) * (1.0f / (float)NSAMP);

    // ---------------- sigma: 8 contiguous samples per lane (coalesced f4) ----
    const float4* sp = (const float4*)(sigmas + (size_t)ray * NSAMP + lane * SPL);
    const float4 sa = sp[0];
    const float4 sb = sp[1];
    float sd[SPL] = { sa.x*delta, sa.y*delta, sa.z*delta, sa.w*delta,
                      sb.x*delta, sb.y*delta, sb.z*delta, sb.w*delta };

    float c[SPL];
    c[0] = sd[0];
#pragma unroll
    for (int k = 1; k < SPL; ++k) c[k] = c[k-1] + sd[k];
    const float tot = c[SPL-1];

    // wave32 inclusive scan of lane totals -> exclusive prefix for this lane
    float x = tot;
#pragma unroll
    for (int off = 1; off < 32; off <<= 1) {
        const float y = __shfl_up(x, off, 32);
        if (lane >= off) x += y;
    }
    const float excl = x - tot;

    float ws[SPL];
#pragma unroll
    for (int k = 0; k < SPL; ++k) {
        const float ce = excl + (k ? c[k-1] : 0.0f);   // exclusive cumsum
        const float T  = expf(-ce);
        const float al = 1.0f - expf(-sd[k]);
        ws[k] = ((T >= 1e-4f) && hit) ? (al * T) : 0.0f;
    }

    // ---------------- consume staged rgbs from LDS ---------------------------
    asm volatile("s_wait_asynccnt 0x0" ::: "memory");

    const float* lf = (const float*)lds_raw;
    const int lb = wv * (WAVE_STRIDE_B / 4) + lane * (LANE_STRIDE_B / 4);
    float r = 0.0f, g = 0.0f, b = 0.0f;
#pragma unroll
    for (int k = 0; k < SPL; ++k) {
        r = fmaf(ws[k], lf[lb + k*3 + 0], r);
        g = fmaf(ws[k], lf[lb + k*3 + 1], g);
        b = fmaf(ws[k], lf[lb + k*3 + 2], b);
    }

    // wave32 reduction
#pragma unroll
    for (int off = 16; off > 0; off >>= 1) {
        r += __shfl_down(r, off, 32);
        g += __shfl_down(g, off, 32);
        b += __shfl_down(b, off, 32);
    }

    if (lane == 0) {
        float* op = out + (size_t)ray * 3;
        op[0] = r; op[1] = g; op[2] = b;
    }
}

extern "C" void kernel_launch(void* const* d_in, const int* in_sizes, int n_in,
                              void* d_out, int out_size, void* d_ws, size_t ws_size,
                              hipStream_t stream) {
    const float* rays      = (const float*)d_in[0];
    const float* sigmas    = (const float*)d_in[1];
    const float* rgbs      = (const float*)d_in[2];
    const float* center    = (const float*)d_in[3];
    const float* half_size = (const float*)d_in[4];
    float* out = (float*)d_out;

    const int n_rays = in_sizes[0] / 6;          // 65536
    const int blocks = (n_rays + WPB - 1) / WPB; // 8192

    ngp_render_kernel<<<blocks, 256, 0, stream>>>(rays, sigmas, rgbs,
                                                  center, half_size,
                                                  out, n_rays);
}